// Decoder_90486370992920
// MI455X (gfx1250) — compile-verified
//
#include <hip/hip_runtime.h>
#include <cstdint>

typedef __attribute__((ext_vector_type(2))) float v2f;
typedef __attribute__((ext_vector_type(4))) float v4f;

#define WAVES_PER_BLOCK 8
#define APW  8        // agents per wave
#define NBUF 4        // LDS quad-buffering, DMA prefetch distance 2

__global__ __launch_bounds__(256)
void decoder_kernel(const float* __restrict__ part,   // [N,64]
                    const float* __restrict__ aact,   // [64]
                    const float* __restrict__ uu,     // [N,64]
                    const float* __restrict__ W,      // [N,2,2]
                    const float* __restrict__ b,      // [N,2]
                    float* __restrict__ out,          // [2N weights][2N actions]
                    int N)
{
    // per wave: NBUF buffers x (64 p + 64 u floats) = 2KB; 8 waves -> 16KB
    __shared__ float smem[WAVES_PER_BLOCK * NBUF * 128];

    const int lane = threadIdx.x & 31;
    const int wave = threadIdx.x >> 5;
    const int base = (blockIdx.x * WAVES_PER_BLOCK + wave) * APW;

    float* wbuf = &smem[wave * (NBUF * 128)];
    // low 32 bits of flat shared pointer == workgroup-relative LDS byte address
    const uint32_t lds0 = (uint32_t)(uintptr_t)wbuf + (uint32_t)lane * 8u;

    // per-lane global byte addresses for agent `base` (8B per lane, 256B rows)
    uint64_t gp = (uint64_t)(uintptr_t)(part + (size_t)base * 64) + (uint64_t)lane * 8u;
    uint64_t gu = (uint64_t)(uintptr_t)(uu   + (size_t)base * 64) + (uint64_t)lane * 8u;

    // CDNA5 async DMA: stage agent (base+i)'s partition row + u row into LDS
    // buffer `bufsel`, non-temporal (read-once stream, keep L2 clean).
    auto stage = [&](int bufsel, int i) {
        uint32_t lp = lds0 + (uint32_t)bufsel * 512u;
        uint32_t lu = lp + 256u;
        uint64_t gpi = gp + (uint64_t)i * 256u;
        uint64_t gui = gu + (uint64_t)i * 256u;
        asm volatile("global_load_async_to_lds_b64 %0, %1, off th:TH_LOAD_NT\n\t"
                     "global_load_async_to_lds_b64 %2, %3, off th:TH_LOAD_NT"
                     :: "v"(lp), "v"(gpi), "v"(lu), "v"(gui)
                     : "memory");
    };

    auto process = [&](int bufsel, int n) {
        const float* pb = wbuf + bufsel * 128;   // 64 p values
        const float* ub = pb + 64;               // 64 u values
        v2f pp = *(const v2f*)(pb + 2 * lane);
        v2f uv = *(const v2f*)(ub + 2 * lane);

        // argmax_m logit(p)+gumbel(u)  ==  argmax_m  p / ((1-p) * (-log2 u))
        // (outer log removed; ln->log2 rescale is argmax-invariant)
        float d0 = (1.0f - pp.x) * (-__builtin_amdgcn_logf(uv.x));
        float d1 = (1.0f - pp.y) * (-__builtin_amdgcn_logf(uv.y));
        float v0 = pp.x * __builtin_amdgcn_rcpf(d0);
        float v1 = pp.y * __builtin_amdgcn_rcpf(d1);

        // Phase 1: value-only wave32 max butterfly (fmax propagates an input
        // bit-exactly; values are positive & non-NaN).
        float m = fmaxf(v0, v1);
        #pragma unroll
        for (int off = 16; off > 0; off >>= 1)
            m = fmaxf(m, __shfl_xor(m, off, 32));

        // Phase 2: recover the first (lowest) index equal to the max via
        // ballots -> pure SALU. Element index = 2*lane (+1). Matches
        // jnp.argmax's first-occurrence tie-break exactly.
        unsigned mask0 = (unsigned)__ballot(v0 == m);
        unsigned mask1 = (unsigned)__ballot(v1 == m);
        int c0 = mask0 ? (__builtin_ctz(mask0) * 2)     : (1 << 20);
        int c1 = mask1 ? (__builtin_ctz(mask1) * 2 + 1) : (1 << 20);
        int bi = min(c0, c1);

        if (lane == 0) {
            float fi = (float)bi;
            float fa = aact[bi];
            v4f w4 = __builtin_nontemporal_load((const v4f*)(W + (size_t)n * 4));
            v2f b2 = __builtin_nontemporal_load((const v2f*)(b + (size_t)n * 2));
            float y0 = fmaf(w4.x, fi, fmaf(w4.y, fa, b2.x));
            float y1 = fmaf(w4.z, fi, fmaf(w4.w, fa, b2.y));
            const float L2E = 1.4426950408889634f;
            float s0 = __builtin_amdgcn_rcpf(1.0f + __builtin_amdgcn_exp2f(-y0 * L2E));
            float s1 = __builtin_amdgcn_rcpf(1.0f + __builtin_amdgcn_exp2f(-y1 * L2E));
            v2f wgt = {s0, s1};
            v2f act = {s0 > 0.0f ? 1.0f : 0.0f, s1 > 0.0f ? 1.0f : 0.0f};
            __builtin_nontemporal_store(wgt, (v2f*)(out + (size_t)n * 2));
            __builtin_nontemporal_store(act, (v2f*)(out + 2 * (size_t)N + (size_t)n * 2));
        }
    };

    if (base + APW <= N) {
        // ---- fast path: uniform, no per-agent guards, depth-2 DMA pipeline ----
        stage(0, 0);
        stage(1, 1);
        #pragma unroll
        for (int i = 0; i < APW; ++i) {
            if (i + 2 < APW) {
                // buffer (i+2)&3 was last read at iter i-2; retire those ds_loads
                // before the DMA may overwrite it
                asm volatile("s_wait_dscnt 0x0" ::: "memory");
                stage((i + 2) & (NBUF - 1), i + 2);
                // 2 stages (4 async ops) may remain in flight; async loads
                // complete in order, so <=4 means stage i's data is in LDS
                asm volatile("s_wait_asynccnt 0x4" ::: "memory");
            } else if (i + 1 < APW) {
                asm volatile("s_wait_asynccnt 0x2" ::: "memory");
            } else {
                asm volatile("s_wait_asynccnt 0x0" ::: "memory");
            }
            process(i & (NBUF - 1), base + i);
        }
    } else if (base < N) {
        // ---- rare tail path: simple serialized per-agent loop ----
        for (int i = 0; i < APW && base + i < N; ++i) {
            stage(0, i);
            asm volatile("s_wait_asynccnt 0x0" ::: "memory");
            process(0, base + i);
            asm volatile("s_wait_dscnt 0x0" ::: "memory");
        }
    }
}

extern "C" void kernel_launch(void* const* d_in, const int* in_sizes, int n_in,
                              void* d_out, int out_size, void* d_ws, size_t ws_size,
                              hipStream_t stream) {
    (void)n_in; (void)out_size; (void)d_ws; (void)ws_size;
    const float* part = (const float*)d_in[0];   // [N,64]
    const float* aact = (const float*)d_in[1];   // [64]
    const float* u    = (const float*)d_in[2];   // [N,64]
    const float* W    = (const float*)d_in[3];   // [N,2,2]
    const float* b    = (const float*)d_in[4];   // [N,2]
    float* out = (float*)d_out;

    int N = in_sizes[0] / 64;
    const int agentsPerBlock = WAVES_PER_BLOCK * APW;   // 64
    int grid = (N + agentsPerBlock - 1) / agentsPerBlock;

    hipLaunchKernelGGL(decoder_kernel, dim3(grid), dim3(256), 0, stream,
                       part, aact, u, W, b, out, N);
}